// MultiHeadSparseGAT_69441031242187
// MI455X (gfx1250) — compile-verified
//
#include <hip/hip_runtime.h>
#include <hip/hip_bf16.h>

#define NN 50000
#define EE 1600000
#define IN_F 128
#define OUT_F 32
#define HH 4
#define HF (HH * OUT_F)          // 128 = output width
#define ALPHA 0.2f
#define ENC_NEG_INF 0x007FFFFFu  // order-preserving encoding of -inf

typedef float v2f __attribute__((ext_vector_type(2)));
typedef float v8f __attribute__((ext_vector_type(8)));

// ---- order-preserving float<->uint map for atomic max on f32 ----
__device__ __forceinline__ unsigned fenc(float f) {
  unsigned u = __float_as_uint(f);
  return (u & 0x80000000u) ? ~u : (u | 0x80000000u);
}
__device__ __forceinline__ float fdec(unsigned u) {
  return (u & 0x80000000u) ? __uint_as_float(u & 0x7fffffffu)
                           : __uint_as_float(~u);
}
__device__ __forceinline__ float lrelu(float x) { return x > 0.f ? x : ALPHA * x; }

// K0: zero output + denom, init m_enc to enc(-inf), build Wcat[128][128]
// Wcat[k][c] = W[c/32][k][c%32]  (heads concatenated along columns)
__global__ __launch_bounds__(256) void gat_init(float* __restrict__ out,
                                                float* __restrict__ denom,
                                                unsigned* __restrict__ m_enc,
                                                const float* __restrict__ W,
                                                float* __restrict__ wcat) {
  int t = blockIdx.x * 256 + threadIdx.x;
  if (t < NN * HF) out[t] = 0.f;
  if (t < NN * HH) { denom[t] = 0.f; m_enc[t] = ENC_NEG_INF; }
  if (t < IN_F * HF) {
    int k = t >> 7, c = t & 127;
    int head = c >> 5, j = c & 31;
    wcat[t] = W[head * (IN_F * OUT_F) + k * OUT_F + j];
  }
}

// K1: H = X[50000,128] @ Wcat[128,128] via V_WMMA_F32_16X16X4_F32.
// Block: 256 threads = 8 waves; block covers 16 rows x 128 cols; each wave
// owns one 16x16 tile, K swept in steps of 4 (32 WMMAs per wave).
#define XS_STRIDE 132  // pad vs LDS bank conflicts
__global__ __launch_bounds__(256) void gat_gemm_wmma(const float* __restrict__ X,
                                                     const float* __restrict__ Wc,
                                                     float* __restrict__ Hout) {
  __shared__ float xs[16 * XS_STRIDE];
  const int row0 = blockIdx.x * 16;
  // coalesced load of the 16x128 X tile
  for (int i = threadIdx.x; i < 16 * IN_F; i += 256) {
    int r = i >> 7, c = i & 127;
    xs[r * XS_STRIDE + c] = X[(row0 + r) * IN_F + c];
  }
  __syncthreads();

  const int wave = threadIdx.x >> 5;
  const int lane = threadIdx.x & 31;
  const int col0 = wave * 16;
  const int m  = lane & 15;   // M index (A) / N index (B,C)
  const int hi = lane >> 4;   // lane half selects K pair / M+8 rows

  v8f c = {};
  for (int k = 0; k < IN_F; k += 4) {
    const int kk = k + (hi << 1);
    v2f a, b;
    // A 16x4 frag: lanes 0-15 -> K = k,k+1 ; lanes 16-31 -> K = k+2,k+3
    a.x = xs[m * XS_STRIDE + kk];
    a.y = xs[m * XS_STRIDE + kk + 1];
    // B 4x16 frag (row striped across lanes), mirrored K split
    b.x = Wc[(kk + 0) * HF + col0 + m];
    b.y = Wc[(kk + 1) * HF + col0 + m];
    c = __builtin_amdgcn_wmma_f32_16x16x4_f32(
        /*neg_a=*/false, a, /*neg_b=*/false, b,
        /*c_mod=*/(short)0, c, /*reuse_a=*/false, /*reuse_b=*/false);
  }

  // C layout: VGPR r -> M=r (lanes 0-15) / M=r+8 (lanes 16-31), N=lane&15
  const int rbase = row0 + (hi ? 8 : 0);
  const int cn = col0 + m;
#pragma unroll
  for (int r = 0; r < 8; ++r)
    Hout[(rbase + r) * HF + cn] = c[r];
}

// K2: per-node per-head attention scores s_src = h . a_src, s_dst = h . a_dst
__global__ __launch_bounds__(256) void gat_scores(const float* __restrict__ Hbuf,
                                                  const float* __restrict__ a_src,
                                                  const float* __restrict__ a_dst,
                                                  float* __restrict__ s_src,
                                                  float* __restrict__ s_dst) {
  int t = blockIdx.x * 256 + threadIdx.x;
  if (t >= NN * HH) return;
  int n = t >> 2, h = t & 3;
  const float* hp = Hbuf + (size_t)n * HF + h * OUT_F;
  const float* as = a_src + h * OUT_F;
  const float* ad = a_dst + h * OUT_F;
  float ss = 0.f, sd = 0.f;
#pragma unroll
  for (int j = 0; j < OUT_F; ++j) {
    float v = hp[j];
    ss += v * as[j];
    sd += v * ad[j];
  }
  s_src[t] = ss;
  s_dst[t] = sd;
}

// K3: segment max over incoming edges (atomic max on encoded f32)
__global__ __launch_bounds__(256) void gat_edge_max(const int* __restrict__ ei,
                                                    const float* __restrict__ s_src,
                                                    const float* __restrict__ s_dst,
                                                    unsigned* __restrict__ m_enc) {
  int t = blockIdx.x * 256 + threadIdx.x;
  if (t >= EE * HH) return;
  int e = t >> 2, h = t & 3;
  int src = ei[e], dst = ei[EE + e];
  float x = lrelu(s_src[src * HH + h] + s_dst[dst * HH + h]);
  atomicMax(&m_enc[dst * HH + h], fenc(x));
}

// K4: denom = segment_sum(exp(e - m))
__global__ __launch_bounds__(256) void gat_edge_denom(const int* __restrict__ ei,
                                                      const float* __restrict__ s_src,
                                                      const float* __restrict__ s_dst,
                                                      const unsigned* __restrict__ m_enc,
                                                      float* __restrict__ denom) {
  int t = blockIdx.x * 256 + threadIdx.x;
  if (t >= EE * HH) return;
  int e = t >> 2, h = t & 3;
  int src = ei[e], dst = ei[EE + e];
  int sh = dst * HH + h;
  float x = lrelu(s_src[src * HH + h] + s_dst[sh]);
  float ex = expf(x - fdec(m_enc[sh]));
  atomicAdd(&denom[sh], ex);
}

// K5: out[dst] += (exp(e-m)/denom) * h[src]  — one thread per (edge, head, feat)
__global__ __launch_bounds__(256) void gat_aggregate(const int* __restrict__ ei,
                                                     const float* __restrict__ s_src,
                                                     const float* __restrict__ s_dst,
                                                     const unsigned* __restrict__ m_enc,
                                                     const float* __restrict__ denom,
                                                     const float* __restrict__ Hbuf,
                                                     float* __restrict__ out) {
  long long t = (long long)blockIdx.x * 256 + threadIdx.x;
  if (t >= (long long)EE * HF) return;
  int e = (int)(t >> 7);
  int r = (int)(t & 127);   // column in [0,128): head = r>>5, feat = r&31
  int h = r >> 5;
  int src = ei[e], dst = ei[EE + e];
  int sh = dst * HH + h;
  float x = lrelu(s_src[src * HH + h] + s_dst[sh]);
  float attn = expf(x - fdec(m_enc[sh])) / denom[sh];
  float val = attn * Hbuf[(size_t)src * HF + r];
  atomicAdd(&out[(size_t)dst * HF + r], val);
}

// K6: ELU in place
__global__ __launch_bounds__(256) void gat_elu(float* __restrict__ out) {
  int t = blockIdx.x * 256 + threadIdx.x;
  if (t >= NN * HF) return;
  float v = out[t];
  out[t] = v > 0.f ? v : expm1f(v);
}

extern "C" void kernel_launch(void* const* d_in, const int* in_sizes, int n_in,
                              void* d_out, int out_size, void* d_ws, size_t ws_size,
                              hipStream_t stream) {
  const float* x      = (const float*)d_in[0];   // [N,128]
  const int*   ei     = (const int*)d_in[1];     // [2,E] (src row, dst row)
  const float* W      = (const float*)d_in[2];   // [H,128,32]
  const float* a_src  = (const float*)d_in[3];   // [H,32]
  const float* a_dst  = (const float*)d_in[4];   // [H,32]
  float*       out    = (float*)d_out;           // [N, H*32]

  // workspace layout
  float*    hbuf  = (float*)d_ws;                     // N*128
  float*    wcat  = hbuf + (size_t)NN * HF;           // 128*128
  float*    ssrc  = wcat + IN_F * HF;                 // N*H
  float*    sdst  = ssrc + (size_t)NN * HH;           // N*H
  unsigned* menc  = (unsigned*)(sdst + (size_t)NN * HH); // N*H
  float*    denom = (float*)(menc + (size_t)NN * HH);    // N*H

  const int B = 256;
  int g_init  = (NN * HF + B - 1) / B;                 // covers all init ranges
  int g_gemm  = NN / 16;                               // 3125 (exact)
  int g_score = (NN * HH + B - 1) / B;
  int g_edge  = (EE * HH + B - 1) / B;
  long long agg_t = (long long)EE * HF;
  int g_agg   = (int)((agg_t + B - 1) / B);
  int g_elu   = (NN * HF + B - 1) / B;

  gat_init<<<g_init, B, 0, stream>>>(out, denom, menc, W, wcat);
  gat_gemm_wmma<<<g_gemm, B, 0, stream>>>(x, wcat, hbuf);
  gat_scores<<<g_score, B, 0, stream>>>(hbuf, a_src, a_dst, ssrc, sdst);
  gat_edge_max<<<g_edge, B, 0, stream>>>(ei, ssrc, sdst, menc);
  gat_edge_denom<<<g_edge, B, 0, stream>>>(ei, ssrc, sdst, menc, denom);
  gat_aggregate<<<g_agg, B, 0, stream>>>(ei, ssrc, sdst, menc, denom, hbuf, out);
  gat_elu<<<g_elu, B, 0, stream>>>(out);
}